// ComplexLinear_17368847745469
// MI455X (gfx1250) — compile-verified
//
#include <hip/hip_runtime.h>

// ---------------------------------------------------------------------------
// ComplexLinear spectral layer for MI455X (gfx1250), wave32 + WMMA f32.
// All heavy stages are batched complex GEMMs on V_WMMA_F32_16X16X4_F32 using
// a 3-multiply (Karatsuba) complex product: P1=ArBr, P2=AiBi, P3=(Ar+Ai)(Br+Bi).
// Uniform-base + 32-bit loop-carried offsets (GVS scale_offset addressing) and
// a structurally unrolled K loop (K % 16 == 0 in every stage) keep the inner
// loop to clause'd loads + WMMA only.
// ---------------------------------------------------------------------------

typedef __attribute__((ext_vector_type(2))) float v2f;
typedef __attribute__((ext_vector_type(8))) float v8f;

#define WMMA_F32(a, b, c) \
  __builtin_amdgcn_wmma_f32_16x16x4_f32(false, (a), false, (b), (short)0, (c), false, false)

// Band: wrapped spatial frequencies 0..33 and 158..191 (68 values), padded to 80.
__device__ __forceinline__ int fq_of(int jb) { return (jb < 34) ? jb : (jb + 124); }

// Exact reproduction of the numpy low-pass mask (fp64, same op order).
__device__ __forceinline__ bool lowmask(int i, int j) {
  double di = (double)i - 95.5, dj = (double)j - 95.5;
  double d = sqrt(di * di + dj * dj);
  double dmax = sqrt(2.0 * 95.5 * 95.5);  // distance at corner (0,0)
  return (d / dmax) < 0.25;
}

// ---------------------------------------------------------------------------
// Generic batched complex GEMM:  C[z] = A[z] * B[z]   (row-major, f32)
// One 16x16 C tile per wave (4 waves / block).
// ACPLX: A has an imaginary part.  CCPLX: store imaginary output.
// ---------------------------------------------------------------------------
template <bool ACPLX, bool CCPLX>
__global__ void __launch_bounds__(128) cgemm_wmma_k(
    int M, int N, int K,
    const float* __restrict__ Are, const float* __restrict__ Aim, int lda, long long bsA,
    const float* __restrict__ Bre, const float* __restrict__ Bim, int ldb, long long bsB,
    float* __restrict__ Cre, float* __restrict__ Cim, int ldc, long long bsC) {
  const int lane = threadIdx.x & 31;
  const int wave = threadIdx.x >> 5;
  const int mtile = blockIdx.y * 4 + wave;
  const int ntile = blockIdx.x;
  if (mtile * 16 >= M) return;  // wave-uniform: EXEC stays all-ones for WMMA

  const long long zb = blockIdx.z;

  // Uniform (SGPR) base pointers; all per-lane addressing via 32-bit offsets.
  const float* __restrict__ aReB = Are + zb * bsA;
  const float* __restrict__ aImB = nullptr;
  if constexpr (ACPLX) aImB = Aim + zb * bsA;
  const float* __restrict__ bReB = Bre + zb * bsB;
  const float* __restrict__ bImB = Bim + zb * bsB;

  // ISA f32 fragment maps (wave32):
  //  A 16x4 : lanes 0-15 -> K=0,1 in v0,v1 ; lanes 16-31 -> K=2,3
  //  B 4x16 : lane = n (+16 for K=2,3), vgpr = K&1
  //  C 16x16: vgpr i -> M=i (lanes 0-15) / M=8+i (lanes 16-31), lane&15 = N
  const int lrow = lane & 15;
  const int khalf = (lane >> 4) * 2;
  const int am = mtile * 16 + lrow;
  const int bn = ntile * 16 + lrow;

  int aoff = am * lda + khalf;  // < 2^31 for all stages
  int boff0 = khalf * ldb + bn;
  int boff1 = boff0 + ldb;      // separate induction var -> GVS addressing
  const int bstep = 4 * ldb;

  v8f p1 = {0.f, 0.f, 0.f, 0.f, 0.f, 0.f, 0.f, 0.f};
  v8f p2 = p1, p3 = p1;

  const int kblocks = K >> 4;  // K % 16 == 0 in every stage
  for (int kk = 0; kk < kblocks; ++kk) {
#pragma unroll
    for (int u = 0; u < 4; ++u) {
      v2f a_re = *(const v2f*)(aReB + aoff);  // b64 (lda, khalf even -> aligned)
      v2f b_re, b_im;
      b_re.x = bReB[boff0];
      b_re.y = bReB[boff1];
      b_im.x = bImB[boff0];
      b_im.y = bImB[boff1];

      p1 = WMMA_F32(a_re, b_re, p1);
      if constexpr (ACPLX) {
        v2f a_im = *(const v2f*)(aImB + aoff);
        p2 = WMMA_F32(a_im, b_im, p2);
        if constexpr (CCPLX) {
          v2f a_s, b_s;  // Karatsuba cross term (VALU adds co-execute with XDL)
          a_s.x = a_re.x + a_im.x;
          a_s.y = a_re.y + a_im.y;
          b_s.x = b_re.x + b_im.x;
          b_s.y = b_re.y + b_im.y;
          p3 = WMMA_F32(a_s, b_s, p3);
        }
      } else {
        if constexpr (CCPLX) p3 = WMMA_F32(a_re, b_im, p3);  // A real: Cim = Ar*Bi
      }
      aoff += 4;
      boff0 += bstep;
      boff1 += bstep;
    }
  }

  // Epilogue: recombine products.
  v8f cre, cim;
  if constexpr (ACPLX) {
#pragma unroll
    for (int i = 0; i < 8; ++i) cre[i] = p1[i] - p2[i];
    if constexpr (CCPLX) {
#pragma unroll
      for (int i = 0; i < 8; ++i) cim[i] = p3[i] - p1[i] - p2[i];
    }
  } else {
    cre = p1;
    if constexpr (CCPLX) cim = p3;
  }

  float* __restrict__ cReB = Cre + zb * bsC;
  float* __restrict__ cImB = nullptr;
  if constexpr (CCPLX) cImB = Cim + zb * bsC;
  const int rbase = mtile * 16 + (lane >> 4) * 8;
#pragma unroll
  for (int i = 0; i < 8; ++i) {
    const int row = rbase + i;
    if (row < M) {  // only the M=8 mixing stage has partial tiles
      const int coff = row * ldc + bn;
      cReB[coff] = cre[i];
      if constexpr (CCPLX) cImB[coff] = cim[i];
    }
  }
}

// ---------------------------------------------------------------------------
// Init kernels: DFT matrices, mask-rank table, kernel scatter.
// ---------------------------------------------------------------------------
__global__ void k_mats(float* d16re, float* d16im, float* eftre, float* eftim,
                       float* g32re, float* g32im, float* ginvre, float* ginvim) {
  int idx = blockIdx.x * blockDim.x + threadIdx.x;
  const double TWO_PI = 6.283185307179586476925286766559;
  if (idx < 256) {  // forward DFT-16 over C:  D16[c][k] = e^{-2pi i ck/16}
    int c = idx >> 4, k = idx & 15;
    double th = TWO_PI * (double)((c * k) & 15) / 16.0;
    d16re[idx] = (float)cos(th);
    d16im[idx] = (float)(-sin(th));
  } else if (idx < 256 + 80 * 192) {  // EfT[jb][x] = e^{-2pi i fq(jb) x/192}, pad rows 0
    int e = idx - 256, jb = e / 192, x = e % 192;
    float re = 0.f, im = 0.f;
    if (jb < 68) {
      int f = fq_of(jb);
      double th = TWO_PI * (double)((f * x) % 192) / 192.0;
      re = (float)cos(th);
      im = (float)(-sin(th));
    }
    eftre[e] = re;
    eftim[e] = im;
  } else if (idx < 256 + 80 * 192 + 1024) {  // G32[f][t] = e^{+2pi i ft/32}/32
    int e = idx - (256 + 80 * 192), f = e >> 5, t = e & 31;
    double th = TWO_PI * (double)((f * t) & 31) / 32.0;
    g32re[e] = (float)(cos(th) / 32.0);
    g32im[e] = (float)(sin(th) / 32.0);
  } else if (idx < 256 + 80 * 192 + 1024 + 192 * 80) {  // Ginv[x][jb] = e^{+2pi i fq x/192}/192
    int e = idx - (256 + 80 * 192 + 1024), x = e / 80, jb = e % 80;
    float re = 0.f, im = 0.f;
    if (jb < 68) {
      int f = fq_of(jb);
      double th = TWO_PI * (double)((f * x) % 192) / 192.0;
      re = (float)(cos(th) / 192.0);
      im = (float)(sin(th) / 192.0);
    }
    ginvre[e] = re;
    ginvim[e] = im;
  }
}

// rank[ib*80+jb] = position of masked cell among row-major flatnonzero order, else -1
__global__ void k_rank(int* __restrict__ rank) {
  int idx = blockIdx.x * blockDim.x + threadIdx.x;
  if (idx >= 80 * 80) return;
  int ib = idx / 80, jb = idx % 80;
  int r = -1;
  if (ib < 68 && jb < 68) {
    int i = (fq_of(ib) + 96) % 192;  // unshift: frequency index -> mask coordinate
    int j = (fq_of(jb) + 96) % 192;
    if (lowmask(i, j)) {
      int cnt = 0;
      for (int i2 = 62; i2 <= i; ++i2) {
        int jend = (i2 == i) ? j : 130;
        for (int j2 = 62; j2 < jend; ++j2)
          if (lowmask(i2, j2)) cnt++;
      }
      r = cnt;
    }
  }
  rank[idx] = r;
}

// Kmat[(ib*80+jb)*512 + c*32 + f] from flat real/imag weights (zero off-mask)
__global__ void k_scatter(const float* __restrict__ rk, const float* __restrict__ ik,
                          const int* __restrict__ rank, float* __restrict__ Kre,
                          float* __restrict__ Kim) {
  int idx = blockIdx.x * blockDim.x + threadIdx.x;
  if (idx >= 80 * 80 * 512) return;
  int p = idx >> 9;
  int cf = idx & 511;
  int r = rank[p];
  float vr = 0.f, vi = 0.f;
  if (r >= 0) {
    long long o = (long long)r * 512 + cf;
    vr = rk[o];
    vi = ik[o];
  }
  Kre[idx] = vr;
  Kim[idx] = vi;
}

// ---------------------------------------------------------------------------
extern "C" void kernel_launch(void* const* d_in, const int* in_sizes, int n_in,
                              void* d_out, int out_size, void* d_ws, size_t ws_size,
                              hipStream_t stream) {
  (void)in_sizes; (void)n_in; (void)out_size; (void)ws_size;
  const float* X = (const float*)d_in[0];   // (8,192,192,16)
  const float* rk = (const float*)d_in[1];  // (1830912,)
  const float* ik = (const float*)d_in[2];
  float* out = (float*)d_out;               // (8,192,192,32)

  char* p = (char*)d_ws;
  auto alloc = [&](size_t nflt) -> float* {
    float* r = (float*)p;
    p += ((nflt * sizeof(float) + 255) / 256) * 256;
    return r;
  };
  // Buffer reuse: bufY holds Y(S1-S2) -> U(S3-S4) -> V2(S5-S6); bufZ holds Z -> V.
  float* bufYre = alloc(4718592);  float* bufYim = alloc(4718592);
  float* bufZre = alloc(1966080);  float* bufZim = alloc(1966080);
  float* bufTre = alloc(3932160);  float* bufTim = alloc(3932160);
  float* Kre = alloc(3276800);     float* Kim = alloc(3276800);
  float* D16re = alloc(256);       float* D16im = alloc(256);
  float* EfTre = alloc(15360);     float* EfTim = alloc(15360);
  float* G32re = alloc(1024);      float* G32im = alloc(1024);
  float* Ginvre = alloc(15360);    float* Ginvim = alloc(15360);
  int* rank = (int*)alloc(6400);

  k_mats<<<dim3((32000 + 255) / 256), dim3(256), 0, stream>>>(
      D16re, D16im, EfTre, EfTim, G32re, G32im, Ginvre, Ginvim);
  k_rank<<<dim3((6400 + 255) / 256), dim3(256), 0, stream>>>(rank);
  k_scatter<<<dim3((3276800 + 255) / 256), dim3(256), 0, stream>>>(rk, ik, rank, Kre, Kim);

  dim3 blk(128, 1, 1);
  // S1: DFT over C.  Y[(b,h,w), k] = X[(b,h,w), c] * D16[c][k]     M=294912 N=16 K=16
  cgemm_wmma_k<false, true><<<dim3(1, 4608, 1), blk, 0, stream>>>(
      294912, 16, 16, X, nullptr, 16, 0, D16re, D16im, 16, 0, bufYre, bufYim, 16, 0);
  // S2: truncated DFT over W (batch (b,h)=1536).  Z[jb][c] = EfT[jb][w] * Y[w][c]
  cgemm_wmma_k<true, true><<<dim3(1, 2, 1536), blk, 0, stream>>>(
      80, 16, 192, EfTre, EfTim, 192, 0, bufYre, bufYim, 16, 3072, bufZre, bufZim, 16, 1280);
  // S3: truncated DFT over H (batch b=8).  U[ib][(jb,c)] = EfT[ib][h] * Z[h][(jb,c)]
  cgemm_wmma_k<true, true><<<dim3(80, 2, 8), blk, 0, stream>>>(
      80, 1280, 192, EfTre, EfTim, 192, 0, bufZre, bufZim, 1280, 245760,
      bufYre, bufYim, 1280, 102400);
  // S4: per-mode C->F mixing (batch (ib,jb)=6400).  V[b][f] = U[b][c] * Kmat[c][f]
  cgemm_wmma_k<true, true><<<dim3(2, 1, 6400), blk, 0, stream>>>(
      8, 32, 16, bufYre, bufYim, 102400, 16, Kre, Kim, 32, 512, bufZre, bufZim, 204800, 32);
  // S5: inverse DFT over F.  V2[r][t] = V[r][f] * G32[f][t]        M=51200 N=32 K=32
  cgemm_wmma_k<true, true><<<dim3(2, 800, 1), blk, 0, stream>>>(
      51200, 32, 32, bufZre, bufZim, 32, 0, G32re, G32im, 32, 0, bufYre, bufYim, 32, 0);
  // S6: inverse over H, band->full (batch b=8).  T[h][(jb,t)] = Ginv[h][ib] * V2[ib][(jb,t)]
  cgemm_wmma_k<true, true><<<dim3(160, 3, 8), blk, 0, stream>>>(
      192, 2560, 80, Ginvre, Ginvim, 80, 0, bufYre, bufYim, 2560, 204800,
      bufTre, bufTim, 2560, 491520);
  // S7: inverse over W, real part only (batch (b,h)=1536) -> d_out
  cgemm_wmma_k<true, false><<<dim3(2, 3, 1536), blk, 0, stream>>>(
      192, 32, 80, Ginvre, Ginvim, 80, 0, bufTre, bufTim, 32, 2560, out, nullptr, 32, 6144);
}